// MV_PPI_Block_12043088298419
// MI455X (gfx1250) — compile-verified
//
#include <hip/hip_runtime.h>
#include <hip/hip_bf16.h>
#include <math.h>

typedef __attribute__((ext_vector_type(2))) float v2f;
typedef __attribute__((ext_vector_type(8))) float v8f;

#define NNODES 6144
#define DIM    128
#define NEDGE  98304
#define NPER   192
#define NB     32
#define NEGS   0.2f

// ---------------------------------------------------------------- utilities
__device__ inline float fast_tanh(float x) {
#if __has_builtin(__builtin_amdgcn_tanhf)
  return __builtin_amdgcn_tanhf(x);
#else
  return tanhf(x);
#endif
}

__device__ inline void atomicMaxFloat(float* addr, float val) {
  // sign-split trick: works for any mix of signs, init = -inf (0xff800000)
  if (val >= 0.0f) atomicMax((int*)addr, __float_as_int(val));
  else             atomicMin((unsigned int*)addr, __float_as_uint(val));
}

__global__ void fill_kernel(float* p, float v, int n) {
  int i = blockIdx.x * blockDim.x + threadIdx.x;
  if (i < n) p[i] = v;
}

__global__ void elu_kernel(const float* __restrict__ x, float* __restrict__ y, int n) {
  int i = blockIdx.x * blockDim.x + threadIdx.x;
  if (i < n) { float v = x[i]; y[i] = v > 0.0f ? v : expm1f(v); }
}

__global__ void bias_add_kernel(float* __restrict__ h, const float* __restrict__ bias, int n) {
  int i = blockIdx.x * blockDim.x + threadIdx.x;
  if (i < n) h[i] += bias[i & (DIM - 1)];
}

// ------------------------------------------------------- fp32 WMMA GEMM
// C[M x 128] = A[M x 128] @ B[128 x 128] (+ bias[col] if bias != null)
// One block = 16 rows x 128 cols; 8 waves, each wave owns a 16x16 tile.
// Uses V_WMMA_F32_16X16X4_F32 (full fp32 precision on the matrix pipe).
__global__ __launch_bounds__(256)
void gemm128_wmma(const float* __restrict__ A, const float* __restrict__ Bm,
                  const float* __restrict__ bias, float* __restrict__ C, int M) {
  __shared__ float sA[16][132];  // pad 128->132 to avoid 64-bank conflicts
  int wave = threadIdx.x >> 5;   // 0..7 -> column tile
  int lane = threadIdx.x & 31;
  int l16  = lane & 15;
  int hi   = lane >> 4;          // 0: K lanes {0,1}, 1: K lanes {2,3}
  int row0 = blockIdx.x * 16;
  int col0 = wave * 16;

  // stage the 16x128 A tile (coalesced global reads)
  for (int t = threadIdx.x; t < 16 * 128; t += 256)
    sA[t >> 7][t & 127] = A[(row0 + (t >> 7)) * DIM + (t & 127)];
  __syncthreads();

  v8f c = {};
  for (int kt = 0; kt < 32; ++kt) {
    int k = kt * 4 + hi * 2;
    v2f a, b;
    // A 16x4 layout: lanes0-15 hold rows, VGPR0/1 = K{0,1}; lanes16-31 = K{2,3}
    a.x = sA[l16][k];
    a.y = sA[l16][k + 1];
    // B 4x16 layout: VGPR v holds row K=v (lanes0-15) / K=v+2 (lanes16-31)
    b.x = Bm[(k    ) * DIM + col0 + l16];
    b.y = Bm[(k + 1) * DIM + col0 + l16];
    c = __builtin_amdgcn_wmma_f32_16x16x4_f32(false, a, false, b, (short)0, c,
                                              false, false);
  }
  float bv = bias ? bias[col0 + l16] : 0.0f;
  int rbase = row0 + (hi << 3);
  for (int v = 0; v < 8; ++v)
    C[(rbase + v) * DIM + col0 + l16] = c[v] + bv;
}

// -------------------------------------------------------------- GAT pieces
// a_src[n,h] = <z[n, h*64:(h+1)*64], att_src[h]>, same for a_dst
__global__ void attcoef_kernel(const float* __restrict__ z,
                               const float* __restrict__ att_src,
                               const float* __restrict__ att_dst,
                               float* __restrict__ asrc, float* __restrict__ adst) {
  int idx = blockIdx.x * blockDim.x + threadIdx.x;
  if (idx >= NNODES * 2) return;
  int n = idx >> 1, h = idx & 1;
  float ss = 0.f, sd = 0.f;
  const float* zr = z + n * DIM + h * 64;
  const float* as = att_src + h * 64;
  const float* ad = att_dst + h * 64;
  for (int f = 0; f < 64; ++f) {
    float zv = zr[f];
    ss += zv * as[f];
    sd += zv * ad[f];
  }
  asrc[idx] = ss;
  adst[idx] = sd;
}

__device__ inline void edge_endpoints(const int* __restrict__ ei, int e,
                                      int& s, int& d) {
  if (e < NEDGE) { s = ei[e]; d = ei[NEDGE + e]; }
  else           { s = d = e - NEDGE; }   // appended self-loops
}

// e = leaky_relu(a_src[src]+a_dst[dst]); segment max over dst (atomic)
__global__ void edge_max_kernel(const int* __restrict__ ei,
                                const float* __restrict__ asrc,
                                const float* __restrict__ adst,
                                float* __restrict__ ebuf, float* __restrict__ emax) {
  int idx = blockIdx.x * blockDim.x + threadIdx.x;
  const int Etot = NEDGE + NNODES;
  if (idx >= Etot * 2) return;
  int e = idx >> 1, h = idx & 1, s, d;
  edge_endpoints(ei, e, s, d);
  float v = asrc[s * 2 + h] + adst[d * 2 + h];
  v = v >= 0.0f ? v : NEGS * v;
  ebuf[idx] = v;
  atomicMaxFloat(&emax[d * 2 + h], v);
}

// ex = exp(e - emax[dst]); segment sum over dst (atomic)
__global__ void edge_exp_kernel(const int* __restrict__ ei,
                                float* __restrict__ ebuf,
                                const float* __restrict__ emax,
                                float* __restrict__ denom) {
  int idx = blockIdx.x * blockDim.x + threadIdx.x;
  const int Etot = NEDGE + NNODES;
  if (idx >= Etot * 2) return;
  int e = idx >> 1, h = idx & 1, s, d;
  edge_endpoints(ei, e, s, d);
  float ex = expf(ebuf[idx] - emax[d * 2 + h]);
  ebuf[idx] = ex;
  atomicAdd(&denom[d * 2 + h], ex);
}

// out[dst,f] += (ex/denom[dst,h]) * z[src,f]    (one thread per edge-feature)
__global__ void edge_aggr_kernel(const int* __restrict__ ei,
                                 const float* __restrict__ ebuf,
                                 const float* __restrict__ denom,
                                 const float* __restrict__ z,
                                 float* __restrict__ out) {
  int idx = blockIdx.x * blockDim.x + threadIdx.x;
  const int Etot = NEDGE + NNODES;
  if (idx >= Etot * DIM) return;
  int e = idx >> 7, f = idx & 127, h = f >> 6, s, d;
  edge_endpoints(ei, e, s, d);
  float al = ebuf[e * 2 + h] / denom[d * 2 + h];
  atomicAdd(&out[d * DIM + f], al * z[s * DIM + f]);
}

// ------------------------------------------------------- mutual attention
// alpha[b,i,j] = sum_d w[d] * tanh(x1[b,i,d] * x2[b,j,d]); 16x16 (i,j) tiles
__global__ __launch_bounds__(256)
void alpha_kernel(const float* __restrict__ x1m, const float* __restrict__ x2m,
                  const float* __restrict__ wv, float* __restrict__ alpha) {
  __shared__ float s1[16][DIM];
  __shared__ float s2[16][DIM];
  __shared__ float sw[DIM];
  int b = blockIdx.z;
  int i0 = blockIdx.x * 16, j0 = blockIdx.y * 16;
  const float* X1 = x1m + (b * NPER + i0) * DIM;
  const float* X2 = x2m + (b * NPER + j0) * DIM;
  for (int t = threadIdx.x; t < 16 * DIM; t += 256) {
    s1[t >> 7][t & 127] = X1[t];
    s2[t >> 7][t & 127] = X2[t];
  }
  if (threadIdx.x < DIM) sw[threadIdx.x] = wv[threadIdx.x];
  __syncthreads();
  int i = threadIdx.x >> 4, j = threadIdx.x & 15;
  float acc = 0.0f;
#pragma unroll 4
  for (int d = 0; d < DIM; ++d)
    acc += sw[d] * fast_tanh(s1[i][d] * s2[j][d]);
  alpha[(b * NPER + i0 + i) * NPER + j0 + j] = acc;
}

// row means (over j) and col means (over i) of alpha, per graph pair
__global__ void mean_kernel(const float* __restrict__ alpha,
                            float* __restrict__ rowm, float* __restrict__ colm) {
  int b = blockIdx.x, t = threadIdx.x;
  if (t >= NPER) return;
  float sr = 0.f, sc = 0.f;
  for (int j = 0; j < NPER; ++j) {
    sr += alpha[(b * NPER + t) * NPER + j];
    sc += alpha[(b * NPER + j) * NPER + t];
  }
  rowm[b * NPER + t] = sr * (1.0f / NPER);
  colm[b * NPER + t] = sc * (1.0f / NPER);
}

// softmax over the 192 scores, then s[b,d] = sum_i x[b,i,d]*p[i]
__global__ __launch_bounds__(256)
void pool_kernel(const float* __restrict__ xm, const float* __restrict__ m,
                 float* __restrict__ out) {
  __shared__ float p[NPER];
  __shared__ float red[256];
  int b = blockIdx.x, t = threadIdx.x;
  float v = (t < NPER) ? m[b * NPER + t] : -INFINITY;
  red[t] = v;
  __syncthreads();
  for (int s = 128; s > 0; s >>= 1) {
    if (t < s) red[t] = fmaxf(red[t], red[t + s]);
    __syncthreads();
  }
  float mx = red[0];
  __syncthreads();
  float e = (t < NPER) ? expf(v - mx) : 0.0f;
  red[t] = e;
  __syncthreads();
  for (int s = 128; s > 0; s >>= 1) {
    if (t < s) red[t] += red[t + s];
    __syncthreads();
  }
  float dn = red[0];
  if (t < NPER) p[t] = e / dn;
  __syncthreads();
  if (t < DIM) {
    float acc = 0.0f;
    for (int i = 0; i < NPER; ++i) acc += xm[(b * NPER + i) * DIM + t] * p[i];
    out[b * DIM + t] = acc;
  }
}

// ------------------------------------------------------------------ driver
extern "C" void kernel_launch(void* const* d_in, const int* in_sizes, int n_in,
                              void* d_out, int out_size, void* d_ws, size_t ws_size,
                              hipStream_t stream) {
  const float* x1       = (const float*)d_in[0];
  const float* x2       = (const float*)d_in[1];
  const int*   ei1      = (const int*)d_in[2];
  const int*   ei2      = (const int*)d_in[3];
  const float* gat_W    = (const float*)d_in[4];
  const float* att_src  = (const float*)d_in[5];
  const float* att_dst  = (const float*)d_in[6];
  const float* gat_bias = (const float*)d_in[7];
  const float* W1       = (const float*)d_in[8];
  const float* b1       = (const float*)d_in[9];
  const float* W2       = (const float*)d_in[10];
  const float* b2       = (const float*)d_in[11];
  const float* w_vec    = (const float*)d_in[12];

  const int NF   = NNODES * DIM;           // 786432
  const int Etot = NEDGE + NNODES;         // 104448

  float* w = (float*)d_ws;
  float* eluX  = w;                 // NF
  float* z     = eluX + NF;         // NF
  float* h1    = z + NF;            // NF
  float* h2    = h1 + NF;           // NF
  float* x1m   = h2 + NF;           // NF
  float* x2m   = x1m + NF;          // NF
  float* asrc  = x2m + NF;          // 12288
  float* adst  = asrc + NNODES * 2; // 12288
  float* emax  = adst + NNODES * 2; // 12288
  float* denom = emax + NNODES * 2; // 12288
  float* ebuf  = denom + NNODES * 2;      // Etot*2 = 208896
  float* alpha = ebuf + Etot * 2;         // 32*192*192 = 1179648
  float* rowm  = alpha + NB * NPER * NPER; // 6144
  float* colm  = rowm + NB * NPER;         // 6144

  const int TB = 256;
  auto blocks = [](int n) { return (n + 255) / 256; };

  // ---- GAT per graph (shared weights), sequential; scratch reused ----
  const float* xin[2] = { x1, x2 };
  const int*   ein[2] = { ei1, ei2 };
  float*       hout[2] = { h1, h2 };
  for (int g = 0; g < 2; ++g) {
    fill_kernel<<<blocks(NF), TB, 0, stream>>>(hout[g], 0.0f, NF);
    fill_kernel<<<blocks(NNODES * 2), TB, 0, stream>>>(emax, -INFINITY, NNODES * 2);
    fill_kernel<<<blocks(NNODES * 2), TB, 0, stream>>>(denom, 0.0f, NNODES * 2);

    elu_kernel<<<blocks(NF), TB, 0, stream>>>(xin[g], eluX, NF);
    gemm128_wmma<<<NNODES / 16, TB, 0, stream>>>(eluX, gat_W, nullptr, z, NNODES);
    attcoef_kernel<<<blocks(NNODES * 2), TB, 0, stream>>>(z, att_src, att_dst,
                                                          asrc, adst);
    edge_max_kernel<<<blocks(Etot * 2), TB, 0, stream>>>(ein[g], asrc, adst,
                                                         ebuf, emax);
    edge_exp_kernel<<<blocks(Etot * 2), TB, 0, stream>>>(ein[g], ebuf, emax, denom);
    edge_aggr_kernel<<<blocks(Etot * DIM), TB, 0, stream>>>(ein[g], ebuf, denom,
                                                            z, hout[g]);
    bias_add_kernel<<<blocks(NF), TB, 0, stream>>>(hout[g], gat_bias, NF);
  }

  // ---- mutual attention ----
  gemm128_wmma<<<NNODES / 16, TB, 0, stream>>>(h1, W1, b1, x1m, NNODES);
  gemm128_wmma<<<NNODES / 16, TB, 0, stream>>>(h2, W2, b2, x2m, NNODES);

  dim3 ag(NPER / 16, NPER / 16, NB);
  alpha_kernel<<<ag, TB, 0, stream>>>(x1m, x2m, w_vec, alpha);

  mean_kernel<<<NB, NPER, 0, stream>>>(alpha, rowm, colm);

  float* out = (float*)d_out;
  pool_kernel<<<NB, TB, 0, stream>>>(x1m, rowm, out);             // s1 [32,128]
  pool_kernel<<<NB, TB, 0, stream>>>(x2m, colm, out + NB * DIM);  // s2 [32,128]
}